// FactorizedSpectralConv3d_88295937671845
// MI455X (gfx1250) — compile-verified
//
#include <hip/hip_runtime.h>

// ============================================================================
// FactorizedSpectralConv3d for MI455X (gfx1250, wave32, WMMA + async-to-LDS).
//
// Per axis:  y = iDFT16 ( W_m ∘ DFT16(x) ),  DFT16 = truncated ortho rfft kept
// to 16 modes => three small GEMM stages entirely in LDS, x read once per axis,
// out written once (y/x passes accumulate). ~19 GFLOP vs ~1 GB HBM traffic:
// memory bound (~44 us @ 23.3 TB/s). FP32 WMMA (v_wmma_f32_16x16x4_f32) keeps
// full reference precision. Slice staging uses global_load_async_to_lds_b128
// (ASYNCcnt) so copies bypass the register file; the z-pass result is written
// with global_store_async_from_lds_b128.
// ============================================================================

typedef __attribute__((ext_vector_type(2))) float v2f;
typedef __attribute__((ext_vector_type(8))) float v8f;

__device__ __forceinline__ v8f wmma_f32_4(v2f a, v2f b, v8f c) {
  // D = A(16x4) * B(4x16) + C(16x16), fp32
  return __builtin_amdgcn_wmma_f32_16x16x4_f32(false, a, false, b, (short)0, c,
                                               false, false);
}

__device__ __forceinline__ unsigned lds_off(const float* p) {
  return (unsigned)(size_t)(const __attribute__((address_space(3))) float*)p;
}

// 16B per lane, global -> LDS, tracked on ASYNCcnt
__device__ __forceinline__ void async_ld_b128(float* lds_dst, const float* g) {
  asm volatile("global_load_async_to_lds_b128 %0, %1, off"
               :: "v"(lds_off(lds_dst)), "v"(g) : "memory");
}
// 16B per lane, LDS -> global, tracked on ASYNCcnt
__device__ __forceinline__ void async_st_b128(float* g, const float* lds_src) {
  asm volatile("global_store_async_from_lds_b128 %0, %1, off"
               :: "v"(g), "v"(lds_off(lds_src)) : "memory");
}
__device__ __forceinline__ void wait_async0() {
  asm volatile("s_wait_asynccnt 0" ::: "memory");
}
__device__ __forceinline__ void wait_ds0() {
  asm volatile("s_wait_dscnt 0" ::: "memory");
}

// LDS layout in floats (dynamic shared memory)
#define OFF_FF 0            // F fwd  [64 n][32 k]                (8 KB)
#define OFF_HI 2048         // H inv  [32 k][64 n]                (8 KB)
#define OFF_XF 4096         // Xf [32 i][32 k][17 p-padded]       (68 KB)
#define OFF_ZB 21504        // Z  [32 o][32 k][17 p-padded]       (68 KB)
#define OFF_XS 38912        // per-wave staging: 8 x 1280 floats  (40 KB)
#define SMEM_FLOATS 49152
#define SMEM_BYTES (SMEM_FLOATS * 4)

__global__ void __launch_bounds__(256)
fno_axis_kernel(const float* __restrict__ x, const float* __restrict__ w,
                float* __restrict__ out,
                int su, int st, long long sn, long long sp, int accumulate) {
  extern __shared__ float sm[];
  float* Ff = sm + OFF_FF;
  float* Hi = sm + OFF_HI;
  float* Xf = sm + OFF_XF;
  float* Zb = sm + OFF_ZB;
  float* Xs = sm + OFF_XS;

  const int tid = threadIdx.x;
  const int lane = tid & 31;
  const int wv = tid >> 5;
  const int lh = lane >> 4;   // lane half (0/1)
  const int lc = lane & 15;   // col/row within half

  // ---- build DFT matrices (ortho rfft, 16 modes) -------------------------
  const float W0 = 6.28318530717958647692f / 64.0f;
  for (int idx = tid; idx < 2048; idx += 256) {  // Ff[n][k], k=2m(+1): Re/Im
    int n = idx >> 5, k = idx & 31, m = k >> 1;
    float s, c;
    __sincosf(W0 * (float)(m * n), &s, &c);
    Ff[idx] = (k & 1) ? (-0.125f * s) : (0.125f * c);
  }
  for (int idx = tid; idx < 2048; idx += 256) {  // Hi[k][n]
    int k = idx >> 6, n = idx & 63, m = k >> 1;
    float amp = (m == 0) ? 0.125f : 0.25f;  // 1/sqrt(64), x2 for m>=1
    float s, c;
    __sincosf(W0 * (float)(m * n), &s, &c);
    Hi[idx] = (k & 1) ? (-amp * s) : (amp * c);  // m=0 imag row -> 0 (sin=0)
  }

  // ---- block geometry: 16-pencil tile ------------------------------------
  const int blk = blockIdx.x;      // 1024 blocks: b(4) x u(64) x t(4)
  const int b = blk >> 8;
  const int u = (blk >> 2) & 63;
  const int t = blk & 3;
  const long long CH = 262144ll;   // channel stride 64^3
  const long long base0 =
      (long long)b * 32ll * CH + (long long)u * su + (long long)t * st;

  __syncthreads();

  // ---- stage 1: forward DFT, all 32 input channels -----------------------
  for (int i = wv; i < 32; i += 8) {
    const float* xi = x + base0 + (long long)i * CH;
    float* xsw = Xs + wv * 1280;
    if (sn == 1) {                // z pass: slice [16 p][68 n]
      for (int e = lane; e < 256; e += 32) {
        int p = e >> 4, n4 = (e & 15) << 2;
        async_ld_b128(xsw + p * 68 + n4, xi + (long long)p * sp + n4);
      }
    } else {                      // y/x pass: slice [64 n][20 p]
      for (int e = lane; e < 256; e += 32) {
        int n = e >> 2, p4 = (e & 3) << 2;
        async_ld_b128(xsw + n * 20 + p4, xi + (long long)n * sn + p4);
      }
    }
    if (i + 8 < 32)
      __builtin_prefetch(x + base0 + (long long)(i + 8) * CH, 0, 3);
    wait_async0();
    __syncthreads();

    v8f a0 = {}, a1 = {};
    if (sn == 1) {
      for (int ks = 0; ks < 16; ++ks) {      // K = 64 over n
        int nb = 4 * ks + 2 * lh;
        v2f av; av.x = xsw[lc * 68 + nb];     av.y = xsw[lc * 68 + nb + 1];
        v2f b0; b0.x = Ff[nb * 32 + lc];      b0.y = Ff[(nb + 1) * 32 + lc];
        v2f b1; b1.x = Ff[nb * 32 + 16 + lc]; b1.y = Ff[(nb + 1) * 32 + 16 + lc];
        a0 = wmma_f32_4(av, b0, a0);
        a1 = wmma_f32_4(av, b1, a1);
      }
    } else {
      for (int ks = 0; ks < 16; ++ks) {
        int nb = 4 * ks + 2 * lh;
        v2f av; av.x = xsw[nb * 20 + lc];     av.y = xsw[(nb + 1) * 20 + lc];
        v2f b0; b0.x = Ff[nb * 32 + lc];      b0.y = Ff[(nb + 1) * 32 + lc];
        v2f b1; b1.x = Ff[nb * 32 + 16 + lc]; b1.y = Ff[(nb + 1) * 32 + 16 + lc];
        a0 = wmma_f32_4(av, b0, a0);
        a1 = wmma_f32_4(av, b1, a1);
      }
    }
    float* dst = Xf + i * 544;               // [k][17]
    const int p0 = 8 * lh;
#pragma unroll
    for (int j = 0; j < 8; ++j) {
      dst[lc * 17 + p0 + j] = a0[j];
      dst[(16 + lc) * 17 + p0 + j] = a1[j];
    }
    __syncthreads();
  }

  // ---- stage 2: per-mode complex channel mix (32i -> 32o) ----------------
  const float2* wc = (const float2*)w;       // w[i][o][m] = (re, im)
  for (int task = wv; task < 32; task += 8) {
    const int m = task >> 1, ot = task & 1;
    v8f zr = {}, zi = {};
    for (int ks = 0; ks < 8; ++ks) {         // K = 32 over i
      int ia = 4 * ks + 2 * lh;
      const float* Xa = Xf + ia * 544 + 2 * m * 17;
      v2f ar; ar.x = Xa[lc];      ar.y = Xa[544 + lc];
      v2f ai; ai.x = Xa[17 + lc]; ai.y = Xa[544 + 17 + lc];
      const int o = ot * 16 + lc;
      float2 wa = wc[(ia * 32 + o) * 16 + m];
      float2 wb = wc[((ia + 1) * 32 + o) * 16 + m];
      v2f wr; wr.x = wa.x;  wr.y = wb.x;
      v2f wi; wi.x = wa.y;  wi.y = wb.y;
      v2f wn; wn.x = -wa.y; wn.y = -wb.y;    // fold minus into B (f32 NEG=0)
      zr = wmma_f32_4(ar, wr, zr);           // Zr += Xr*Wr
      zr = wmma_f32_4(ai, wn, zr);           // Zr -= Xi*Wi
      zi = wmma_f32_4(ar, wi, zi);           // Zi += Xr*Wi
      zi = wmma_f32_4(ai, wr, zi);           // Zi += Xi*Wr
    }
    float* dz = Zb + (ot * 16 + lc) * 544 + 2 * m * 17;
    const int p0 = 8 * lh;
#pragma unroll
    for (int j = 0; j < 8; ++j) {
      dz[p0 + j] = zr[j];
      dz[17 + p0 + j] = zi[j];
    }
  }
  __syncthreads();

  // ---- stage 3: inverse DFT + coalesced (accumulating) global write ------
  for (int o = wv; o < 32; o += 8) {
    v8f c0 = {}, c1 = {}, c2 = {}, c3 = {};
    for (int ks = 0; ks < 8; ++ks) {         // K = 32 over spectral k
      int ka = 4 * ks + 2 * lh;
      const float* Za = Zb + o * 544 + ka * 17;
      v2f av; av.x = Za[lc]; av.y = Za[17 + lc];
      v2f bb;
      bb.x = Hi[ka * 64 + lc];      bb.y = Hi[(ka + 1) * 64 + lc];
      c0 = wmma_f32_4(av, bb, c0);
      bb.x = Hi[ka * 64 + 16 + lc]; bb.y = Hi[(ka + 1) * 64 + 16 + lc];
      c1 = wmma_f32_4(av, bb, c1);
      bb.x = Hi[ka * 64 + 32 + lc]; bb.y = Hi[(ka + 1) * 64 + 32 + lc];
      c2 = wmma_f32_4(av, bb, c2);
      bb.x = Hi[ka * 64 + 48 + lc]; bb.y = Hi[(ka + 1) * 64 + 48 + lc];
      c3 = wmma_f32_4(av, bb, c3);
    }
    float* zt = Xs + wv * 1280;              // transpose buffer [16 p][68 n]
    const int p0 = 8 * lh;
#pragma unroll
    for (int j = 0; j < 8; ++j) {
      zt[(p0 + j) * 68 + lc] = c0[j];
      zt[(p0 + j) * 68 + 16 + lc] = c1[j];
      zt[(p0 + j) * 68 + 32 + lc] = c2[j];
      zt[(p0 + j) * 68 + 48 + lc] = c3[j];
    }
    __syncthreads();
    float* op = out + base0 + (long long)o * CH;
    if (sn == 1) {
      if (!accumulate) {
        // pure write-out: async LDS -> global, no VGPR round trip
        wait_ds0();                          // LDS committed before async read
        for (int e = lane; e < 256; e += 32) {
          int p = e >> 4, n4 = (e & 15) << 2;
          async_st_b128(op + (long long)p * sp + n4, zt + p * 68 + n4);
        }
        wait_async0();                       // before zt is reused
      } else {
        for (int e = lane; e < 256; e += 32) {
          int p = e >> 4, n4 = (e & 15) << 2;
          const float* zz = zt + p * 68 + n4;
          float4 v; v.x = zz[0]; v.y = zz[1]; v.z = zz[2]; v.w = zz[3];
          float4* g = (float4*)(op + (long long)p * sp + n4);
          float4 oo = *g;
          v.x += oo.x; v.y += oo.y; v.z += oo.z; v.w += oo.w;
          *g = v;
        }
      }
    } else {
      for (int e = lane; e < 256; e += 32) {
        int n = e >> 2, p4 = (e & 3) << 2;
        float4 v;
        v.x = zt[(p4 + 0) * 68 + n];
        v.y = zt[(p4 + 1) * 68 + n];
        v.z = zt[(p4 + 2) * 68 + n];
        v.w = zt[(p4 + 3) * 68 + n];
        float4* g = (float4*)(op + (long long)n * sn + p4);
        if (accumulate) {
          float4 oo = *g;
          v.x += oo.x; v.y += oo.y; v.z += oo.z; v.w += oo.w;
        }
        *g = v;
      }
    }
    __syncthreads();
  }
}

extern "C" void kernel_launch(void* const* d_in, const int* in_sizes, int n_in,
                              void* d_out, int out_size, void* d_ws,
                              size_t ws_size, hipStream_t stream) {
  (void)in_sizes; (void)n_in; (void)out_size; (void)d_ws; (void)ws_size;
  const float* x  = (const float*)d_in[0];
  const float* w0 = (const float*)d_in[1];  // axis 2 (x-dim)
  const float* w1 = (const float*)d_in[2];  // axis 3 (y-dim)
  const float* w2 = (const float*)d_in[3];  // axis 4 (z-dim)
  float* out = (float*)d_out;

  (void)hipFuncSetAttribute((const void*)fno_axis_kernel,
                            hipFuncAttributeMaxDynamicSharedMemorySize,
                            SMEM_BYTES);

  dim3 grid(1024), block(256);
  // axis z: pencils along z (sn=1), 16 pencils over y (sp=64); write out
  fno_axis_kernel<<<grid, block, SMEM_BYTES, stream>>>(
      x, w2, out, /*su=*/4096, /*st=*/1024, /*sn=*/1ll, /*sp=*/64ll, 0);
  // axis y: transform along y (sn=64), pencil tile along z (sp=1); accumulate
  fno_axis_kernel<<<grid, block, SMEM_BYTES, stream>>>(
      x, w1, out, /*su=*/4096, /*st=*/16, /*sn=*/64ll, /*sp=*/1ll, 1);
  // axis x: transform along x (sn=4096), pencil tile along z (sp=1); accumulate
  fno_axis_kernel<<<grid, block, SMEM_BYTES, stream>>>(
      x, w0, out, /*su=*/64, /*st=*/16, /*sn=*/4096ll, /*sp=*/1ll, 1);
}